// SSSD_3753801417523
// MI455X (gfx1250) — compile-verified
//
#include <hip/hip_runtime.h>
#include <hip/hip_bf16.h>

#define LSEQ 4096
#define NB   16

typedef __attribute__((ext_vector_type(2))) float v2f;
typedef __attribute__((ext_vector_type(8))) float v8f;
typedef int vi4 __attribute__((vector_size(16)));

#define AS1 __attribute__((address_space(1)))
#define AS3 __attribute__((address_space(3)))

// ---------------------------------------------------------------------------
// Time embedding: te1 = relu(emb(steps) @ Wt1^T + bt1)   (B=16, 128 -> 512)
// ---------------------------------------------------------------------------
__global__ void k_temb1(const int* __restrict__ steps, const float* __restrict__ Wt1,
                        const float* __restrict__ bt1, float* __restrict__ te1) {
  __shared__ float emb[128];
  int b = blockIdx.x, t = threadIdx.x;          // t in [0,512)
  if (t < 128) {
    float st = (float)steps[b];
    int j = t & 63;
    float f = expf(-logf(10000.0f) * (float)j / 63.0f);
    float e = st * f;
    emb[t] = (t < 64) ? sinf(e) : cosf(e);
  }
  __syncthreads();
  float s = bt1[t];
  for (int k = 0; k < 128; ++k) s = fmaf(emb[k], Wt1[t * 128 + k], s);
  te1[b * 512 + t] = fmaxf(s, 0.0f);
}

// te = relu(te1 @ Wt2^T + bt2)   (512 -> 512)
__global__ void k_temb2(const float* __restrict__ te1, const float* __restrict__ Wt2,
                        const float* __restrict__ bt2, float* __restrict__ te) {
  __shared__ float r[512];
  int b = blockIdx.x, t = threadIdx.x;
  r[t] = te1[b * 512 + t];
  __syncthreads();
  float s = bt2[t];
  for (int k = 0; k < 512; ++k) s = fmaf(r[k], Wt2[t * 512 + k], s);
  te[b * 512 + t] = fmaxf(s, 0.0f);
}

// teb[b][d] = te[b] . Wt[d] + bt[d]   (per-resblock projection, block = D threads)
__global__ void k_teb(const float* __restrict__ te, const float* __restrict__ Wt,
                      const float* __restrict__ bt, float* __restrict__ teb, int D) {
  extern __shared__ float r[];
  int b = blockIdx.x, t = threadIdx.x;
  for (int k = t; k < 512; k += blockDim.x) r[k] = te[b * 512 + k];
  __syncthreads();
  float s = bt[t];
  for (int k = 0; k < 512; ++k) s = fmaf(r[k], Wt[t * 512 + k], s);
  teb[b * D + t] = s;
}

// ---------------------------------------------------------------------------
// cond2 prep: c0 = cond*mask, c1 = mask
// ---------------------------------------------------------------------------
__global__ void k_cond2(const float* __restrict__ cond, const float* __restrict__ mask,
                        float* __restrict__ c0, float* __restrict__ c1, int n) {
  int i = blockIdx.x * blockDim.x + threadIdx.x;
  if (i < n) { float m = mask[i]; c0[i] = cond[i] * m; c1[i] = m; }
}

// x = relu(Winit[d]*noise + binit[d])  over (B,128,L)
__global__ void k_init(const float* __restrict__ noise, const float* __restrict__ Wi,
                       const float* __restrict__ bi, float* __restrict__ x) {
  int i = blockIdx.x * blockDim.x + threadIdx.x;
  int l = i & (LSEQ - 1);
  int d = (i >> 12) & 127;
  int b = i >> 19;
  float v = fmaf(Wi[d], noise[b * LSEQ + l], bi[d]);
  x[i] = fmaxf(v, 0.0f);
}

__global__ void k_add(float* __restrict__ y, const float* __restrict__ a, int n) {
  int i = blockIdx.x * blockDim.x + threadIdx.x;
  if (i < n) y[i] += a[i];
}

// ---------------------------------------------------------------------------
// LayerNorm over channel dim D at each (b,l); optional per-(b,d) te add.
// grid (L/256, B), block 256; thread owns one l, loops channels (coalesced).
// Per-channel te/gamma/beta staged once in LDS.
// ---------------------------------------------------------------------------
__global__ void k_ln(const float* __restrict__ X, const float* __restrict__ teb,
                     const float* __restrict__ g, const float* __restrict__ bb,
                     float* __restrict__ Z, int D) {
  extern __shared__ float smln[];         // [3*D] : tadd | gamma | beta
  float* tad = smln;
  float* gg  = smln + D;
  float* bbx = smln + 2 * D;
  int b = blockIdx.y;
  for (int d = threadIdx.x; d < D; d += blockDim.x) {
    tad[d] = teb ? teb[b * D + d] : 0.0f;
    gg[d]  = g[d];
    bbx[d] = bb[d];
  }
  __syncthreads();
  int l = blockIdx.x * blockDim.x + threadIdx.x;
  const float* xp = X + (size_t)b * D * LSEQ + l;
  float s = 0.f, ss = 0.f;
  for (int d = 0; d < D; ++d) {
    float v = xp[(size_t)d * LSEQ] + tad[d];
    s += v; ss = fmaf(v, v, ss);
  }
  float mean = s / (float)D;
  float var  = ss / (float)D - mean * mean;
  float rstd = rsqrtf(var + 1e-5f);
  float* zp = Z + (size_t)b * D * LSEQ + l;
  for (int d = 0; d < D; ++d) {
    float v = xp[(size_t)d * LSEQ] + tad[d];
    zp[(size_t)d * LSEQ] = fmaf((v - mean) * rstd, gg[d], bbx[d]);
  }
}

// ---------------------------------------------------------------------------
// S4 diagonal recurrence: one wave per (b,d). 64 states = 2 per lane.
//   h = tanh(A*h + B*x_t);  y_t = sum_n C*h + Dv*x_t
// x broadcast from a 32-wide register chunk, butterfly-reduce C.h each step.
// ---------------------------------------------------------------------------
__global__ __launch_bounds__(32)
void k_s4(const float* __restrict__ Z, float* __restrict__ Y,
          const float* __restrict__ Ad, const float* __restrict__ Bm,
          const float* __restrict__ Cm, const float* __restrict__ Dv, int D) {
  int bd = blockIdx.x;           // = b*D + d
  int d  = bd % D;
  int lane = threadIdx.x;
  float a0 = Ad[d * 64 + lane],      a1 = Ad[d * 64 + 32 + lane];
  float b0 = Bm[d * 64 + lane],      b1 = Bm[d * 64 + 32 + lane];
  float c0 = Cm[d * 64 + lane],      c1 = Cm[d * 64 + 32 + lane];
  float dv = Dv[d];
  const float* zp = Z + (size_t)bd * LSEQ;
  float*       yp = Y + (size_t)bd * LSEQ;
  float h0 = 0.f, h1 = 0.f;
  for (int t0 = 0; t0 < LSEQ; t0 += 32) {
    float xch = zp[t0 + lane];
    if (t0 + 32 < LSEQ) __builtin_prefetch(zp + t0 + 32 + lane, 0, 1);
    float yacc = 0.f;
    #pragma unroll
    for (int tt = 0; tt < 32; ++tt) {
      float xt = __shfl(xch, tt, 32);
      h0 = tanhf(fmaf(a0, h0, b0 * xt));
      h1 = tanhf(fmaf(a1, h1, b1 * xt));
      float s = fmaf(c0, h0, c1 * h1);
      s += __shfl_xor(s, 16, 32);
      s += __shfl_xor(s,  8, 32);
      s += __shfl_xor(s,  4, 32);
      s += __shfl_xor(s,  2, 32);
      s += __shfl_xor(s,  1, 32);
      if (lane == tt) yacc = fmaf(dv, xt, s);
    }
    yp[t0 + lane] = yacc;
  }
}

// ---------------------------------------------------------------------------
// fp32 WMMA GEMM: Y[b,o,l] = sum_c W[o,c]*X[b,c,l] + bias + (cond) + (resid).
// Block = 256 threads = 8 waves covering M = 128 rows x N = 16 cols.
// The C x 16 X-tile is staged ONCE into LDS via GLOBAL_LOAD_ASYNC_TO_LDS_B128
// (ASYNCcnt-tracked), then all 8 waves consume it -> 8x less global X traffic.
// A layout per ISA 7.12.2: lanes 0-15 rows K=0,1 ; lanes 16-31 rows K=2,3.
// C/D layout: VGPR r -> rows (r, r+8), cols = lane&15.
// ---------------------------------------------------------------------------
__global__ __launch_bounds__(256)
void k_gemm(const float* __restrict__ W, const float* __restrict__ bias,
            const float* __restrict__ X, float* __restrict__ Y,
            const float* __restrict__ resid,
            const float* __restrict__ Wc, const float* __restrict__ bc,
            const float* __restrict__ c0, const float* __restrict__ c1,
            int O, int C, int relu) {
  extern __shared__ float lds[];          // [C*16] : X tile, row-major [k][n]
  int b    = blockIdx.z;
  int l0   = blockIdx.x * 16;
  int mb   = blockIdx.y * 128;            // M block base
  int tid  = threadIdx.x;
  const float* Xb = X + (size_t)b * C * LSEQ;

  // ---- stage X[k=0..C-1][n=0..15] tile into LDS, 16B chunks per thread ----
  int nchunk = C * 4;                     // (C*16 floats)/4 per chunk
  for (int q = tid; q < nchunk; q += 256) {
    int k = q >> 2, c4 = (q & 3) * 4;
    const float* gp = Xb + (size_t)k * LSEQ + l0 + c4;
#if __has_builtin(__builtin_amdgcn_global_load_async_to_lds_b128)
    __builtin_amdgcn_global_load_async_to_lds_b128(
        (AS1 vi4*)gp, (AS3 vi4*)&lds[q * 4], 0, 0);
#else
    *(float4*)&lds[q * 4] = *(const float4*)gp;
#endif
  }
#if __has_builtin(__builtin_amdgcn_global_load_async_to_lds_b128)
  asm volatile("s_wait_asynccnt 0" ::: "memory");
#endif
  __syncthreads();

  // ---- 8 waves, one 16x16 WMMA tile each over M ----
  int wid  = tid >> 5;                    // wave id 0..7
  int lane = tid & 31;
  int half = lane >> 4;                   // 0: K=0,1   1: K=2,3
  int lid  = lane & 15;
  int m0   = mb + wid * 16;

  v8f acc = {};
  for (int k = 0; k < C; k += 4) {
    int ka = k + half * 2;
    v2f a, bm;
    a.x  = W[(m0 + lid) * C + ka];
    a.y  = W[(m0 + lid) * C + ka + 1];
    bm.x = lds[ka * 16 + lid];
    bm.y = lds[(ka + 1) * 16 + lid];
    acc = __builtin_amdgcn_wmma_f32_16x16x4_f32(false, a, false, bm,
                                                (short)0, acc, false, false);
  }
  int l = l0 + lid;
  float cv0 = 0.f, cv1 = 0.f;
  if (Wc) { cv0 = c0[(size_t)b * LSEQ + l]; cv1 = c1[(size_t)b * LSEQ + l]; }
  #pragma unroll
  for (int r = 0; r < 8; ++r) {
    int m = m0 + r + half * 8;
    float v = acc[r] + bias[m];
    if (Wc)    v += Wc[m * 2 + 0] * cv0 + Wc[m * 2 + 1] * cv1 + bc[m];
    if (resid) v += resid[((size_t)b * O + m) * LSEQ + l];
    if (relu)  v = fmaxf(v, 0.0f);
    Y[((size_t)b * O + m) * LSEQ + l] = v;
  }
}

// Final head matvec: out[b,l] = bf2 + sum_d Wf2[d]*X[b,d,l]
__global__ void k_out(const float* __restrict__ X, const float* __restrict__ w,
                      const float* __restrict__ bias, float* __restrict__ out) {
  int b = blockIdx.y;
  int l = blockIdx.x * blockDim.x + threadIdx.x;
  float s = bias[0];
  for (int d = 0; d < 128; ++d)
    s = fmaf(w[d], X[((size_t)b * 128 + d) * LSEQ + l], s);
  out[(size_t)b * LSEQ + l] = s;
}

// ===========================================================================
extern "C" void kernel_launch(void* const* d_in, const int* in_sizes, int n_in,
                              void* d_out, int out_size, void* d_ws, size_t ws_size,
                              hipStream_t stream) {
  (void)in_sizes; (void)n_in; (void)out_size; (void)ws_size;
  // ---- input leaf indexing (setup_inputs insertion order, depth-first) ----
  // 0 noise, 1 cond, 2 mask, 3 steps, 4 Winit, 5 binit,
  // 6..53 rb_down[0..3] (12 each: Adiag,B,C,D,Wp,bp,ln_g,ln_b,Wt,bt,Wc,bc),
  // 54 Wdown, 55 bdown, 56..103 rb_mid, 104 Wup, 105 bup, 106..153 rb_up,
  // 154 norm_g, 155 norm_b, 156 Wf1, 157 bf1, 158 Wf2, 159 bf2,
  // 160 Wt1, 161 bt1, 162 Wt2, 163 bt2
  auto F = [&](int i) -> const float* { return (const float*)d_in[i]; };

  const float* noise = F(0);
  const float* cond  = F(1);
  const float* mask  = F(2);
  const int*   steps = (const int*)d_in[3];

  // ---- workspace layout ----
  char* w = (char*)d_ws;
  float* bufA  = (float*)(w);                       // 64 MiB
  float* bufB  = (float*)(w + (64ull  << 20));      // 64 MiB
  float* bufC  = (float*)(w + (128ull << 20));      // 64 MiB
  float* saveD = (float*)(w + (192ull << 20));      // 64 MiB (H-channel skip)
  float* save4 = (float*)(w + (256ull << 20));      // 32 MiB (D0-channel skip)
  char* sm = w + (288ull << 20);
  float* te1 = (float*)sm; sm += 16 * 512 * 4;
  float* te  = (float*)sm; sm += 16 * 512 * 4;
  float* teb = (float*)sm; sm += 16 * 256 * 4;
  float* c0  = (float*)sm; sm += 16 * LSEQ * 4;
  float* c1  = (float*)sm; sm += 16 * LSEQ * 4;

  // ---- time embedding + cond prep ----
  k_temb1<<<NB, 512, 0, stream>>>(steps, F(160), F(161), te1);
  k_temb2<<<NB, 512, 0, stream>>>(te1, F(162), F(163), te);
  k_cond2<<<(NB * LSEQ) / 256, 256, 0, stream>>>(cond, mask, c0, c1, NB * LSEQ);

  float* x  = bufA;
  float* zb = bufB;
  float* yb = bufC;
  k_init<<<(NB * 128 * LSEQ) / 256, 256, 0, stream>>>(noise, F(4), F(5), x);

  auto run_gemm = [&](const float* Wm, const float* bi, const float* Xi, float* Yo,
                      const float* resid, const float* Wc, const float* bcv,
                      int O, int C, int relu) {
    k_gemm<<<dim3(LSEQ / 16, O / 128, NB), 256, C * 16 * 4, stream>>>(
        Wm, bi, Xi, Yo, resid, Wc, bcv, c0, c1, O, C, relu);
  };

  auto run_rb = [&](int base, int D) {
    k_teb<<<NB, D, 512 * 4, stream>>>(te, F(base + 8), F(base + 9), teb, D);
    k_ln<<<dim3(LSEQ / 256, NB), 256, 3 * D * 4, stream>>>(
        x, teb, F(base + 6), F(base + 7), zb, D);
    k_s4<<<NB * D, 32, 0, stream>>>(zb, yb, F(base + 0), F(base + 1),
                                    F(base + 2), F(base + 3), D);
    run_gemm(F(base + 4), F(base + 5), yb, zb, x, F(base + 10), F(base + 11), D, D, 0);
    float* t = x; x = zb; zb = t;   // new x is the GEMM output
  };

  // ---- down blocks (D=128) ----
  for (int i = 0; i < 4; ++i) run_rb(6 + 12 * i, 128);
  (void)hipMemcpyAsync(save4, x, (size_t)NB * 128 * LSEQ * 4,
                       hipMemcpyDeviceToDevice, stream);

  // ---- downsample: H = relu(Wdown @ x) ----
  run_gemm(F(54), F(55), x, zb, nullptr, nullptr, nullptr, 256, 128, 1);
  { float* t = x; x = zb; zb = t; }
  (void)hipMemcpyAsync(saveD, x, (size_t)NB * 256 * LSEQ * 4,
                       hipMemcpyDeviceToDevice, stream);

  // ---- mid blocks (D=256) ----
  for (int i = 0; i < 4; ++i) run_rb(56 + 12 * i, 256);
  k_add<<<(NB * 256 * LSEQ) / 256, 256, 0, stream>>>(x, saveD, NB * 256 * LSEQ);

  // ---- upsample: x = Wup @ x (no relu) ----
  run_gemm(F(104), F(105), x, zb, nullptr, nullptr, nullptr, 128, 256, 0);
  { float* t = x; x = zb; zb = t; }

  // ---- up blocks (D=128) ----
  for (int i = 0; i < 4; ++i) run_rb(106 + 12 * i, 128);
  k_add<<<(NB * 128 * LSEQ) / 256, 256, 0, stream>>>(x, save4, NB * 128 * LSEQ);

  // ---- head: LN -> relu(Wf1) -> Wf2 matvec ----
  k_ln<<<dim3(LSEQ / 256, NB), 256, 3 * 128 * 4, stream>>>(
      x, nullptr, F(154), F(155), zb, 128);
  run_gemm(F(156), F(157), zb, yb, nullptr, nullptr, nullptr, 128, 128, 1);
  k_out<<<dim3(LSEQ / 256, NB), 256, 0, stream>>>(yb, F(158), F(159), (float*)d_out);
}